// CollectAtomTriples_47906065219823
// MI455X (gfx1250) — compile-verified
//
#include <hip/hip_runtime.h>
#include <stdint.h>

// Native clang vector types so __builtin_nontemporal_store lowers to
// global_store_b128 with the NT temporal hint (write-once output, 297MB >
// 192MB L2 -> keep L2 clean).
typedef int      v4i __attribute__((ext_vector_type(4)));
typedef uint32_t v4u __attribute__((ext_vector_type(4)));

// ---------------------------------------------------------------------------
// Kernel 1: segment-start detection, equivalent to bincount+exclusive-cumsum
// on a sorted idx_i. base[atom] = first row index of that atom's segment.
// 6.4 MB of reads; negligible next to the 297 MB of output stores.
// ---------------------------------------------------------------------------
__global__ void seg_base_kernel(const int* __restrict__ idx_i, int n_total,
                                int* __restrict__ base) {
    int t = blockIdx.x * blockDim.x + threadIdx.x;
    if (t >= n_total) return;
    int a = idx_i[t];
    if (t == 0 || idx_i[t - 1] != a) base[a] = t;
}

// ---------------------------------------------------------------------------
// Kernel 2: triple generation. One thread produces 4 consecutive triples
// (one 128-bit store into each of the three output arrays).
//
// The (jj,kk) pair table (P entries, packed (j<<16)|k) is built once per
// workgroup into LDS and read back 4-at-a-time with ds_load_b128; the hot
// path per thread is then 3 vector adds + 3 global_store_b128 (NT).
//
// QP_CONST > 0 specializes the quads-per-atom divisor at compile time so the
// a = t / QP division becomes a mulhi sequence.
// ---------------------------------------------------------------------------
template <int QP_CONST>
__global__ void triples_quad_kernel(const int* __restrict__ base,
                                    int* __restrict__ out_i,
                                    int* __restrict__ out_j,
                                    int* __restrict__ out_k,
                                    int K, int P, int QP_rt, int nquad_total) {
    __shared__ __align__(16) uint32_t s_pairs[1024];  // 4KB LDS, P <= 1024

    // Build the shared pair table (row-major triu order, matches
    // torch.combinations / jnp.triu_indices(K, k=1)).
    for (int p = threadIdx.x; p < P; p += blockDim.x) {
        int pp = p, j = 0, row = K - 1;
        while (pp >= row) { pp -= row; --row; ++j; }
        int k = j + 1 + pp;
        s_pairs[p] = ((uint32_t)j << 16) | (uint32_t)k;
    }
    __syncthreads();

    int t = blockIdx.x * blockDim.x + threadIdx.x;
    if (t >= nquad_total) return;

    const int QP = (QP_CONST > 0) ? QP_CONST : QP_rt;   // quads per atom = P/4
    int a  = t / QP;
    int q  = t - a * QP;
    int p0 = q * 4;

    int base_a = base ? base[a] : a * K;   // uniform branch; base[a] L2-cached

    // 16B-aligned LDS read of 4 packed pairs (p0 is a multiple of 4).
    v4u pk = *reinterpret_cast<const v4u*>(&s_pairs[p0]);

    v4i vi; vi.x = a; vi.y = a; vi.z = a; vi.w = a;
    v4i vj; v4i vk;
    vj.x = base_a + (int)(pk.x >> 16);   vk.x = base_a + (int)(pk.x & 0xFFFFu);
    vj.y = base_a + (int)(pk.y >> 16);   vk.y = base_a + (int)(pk.y & 0xFFFFu);
    vj.z = base_a + (int)(pk.z >> 16);   vk.z = base_a + (int)(pk.z & 0xFFFFu);
    vj.w = base_a + (int)(pk.w >> 16);   vk.w = base_a + (int)(pk.w & 0xFFFFu);

    size_t off = (size_t)a * (size_t)P + (size_t)p0;    // multiple of 4 -> 16B aligned
    __builtin_nontemporal_store(vi, reinterpret_cast<v4i*>(out_i + off));
    __builtin_nontemporal_store(vj, reinterpret_cast<v4i*>(out_j + off));
    __builtin_nontemporal_store(vk, reinterpret_cast<v4i*>(out_k + off));
}

// Scalar remainder path for P % 4 != 0 (unused for K=32 where P=496).
__global__ void triples_rem_kernel(const int* __restrict__ base,
                                   int* __restrict__ out_i,
                                   int* __restrict__ out_j,
                                   int* __restrict__ out_k,
                                   int K, int P, int p_start, int n_atoms) {
    int rem = P - p_start;
    int t = blockIdx.x * blockDim.x + threadIdx.x;
    if (t >= n_atoms * rem) return;
    int a = t / rem;
    int p = p_start + (t - a * rem);

    int pp = p, j = 0, row = K - 1;
    while (pp >= row) { pp -= row; --row; ++j; }
    int k = j + 1 + pp;

    int base_a = base ? base[a] : a * K;
    size_t off = (size_t)a * (size_t)P + (size_t)p;
    out_i[off] = a;
    out_j[off] = base_a + j;
    out_k[off] = base_a + k;
}

// ---------------------------------------------------------------------------
// Host launcher. Derives (N, K, P) from in_sizes[0] = N*K and
// out_size = 3*N*P = 3*n_total*(K-1)/2, which is strictly increasing in K,
// so the solution is unique. No device reads -> graph-capture safe.
// ---------------------------------------------------------------------------
extern "C" void kernel_launch(void* const* d_in, const int* in_sizes, int n_in,
                              void* d_out, int out_size, void* d_ws, size_t ws_size,
                              hipStream_t stream) {
    const int* idx_i = (const int*)d_in[0];
    const long long n_total = in_sizes[0];            // N * K

    long long K = 0, N = 0, P = 0;
    for (long long k = 2; k <= 4096; ++k) {
        if (n_total % k) continue;
        long long n = n_total / k;
        long long p = k * (k - 1) / 2;
        if (3LL * n * p == (long long)out_size) { K = k; N = n; P = p; break; }
    }
    if (K == 0) { K = 32; N = n_total / 32; P = 496; }  // problem default

    int* out   = (int*)d_out;
    int* out_i = out;
    int* out_j = out + (size_t)N * (size_t)P;
    int* out_k = out + 2 * (size_t)N * (size_t)P;

    // base[] from the actual idx_i input (reference-faithful); fall back to
    // the analytic a*K layout if the scratch buffer can't hold it.
    int* base = nullptr;
    if (ws_size >= (size_t)N * sizeof(int)) {
        base = (int*)d_ws;
        int threads = 256;
        int blocks  = (int)((n_total + threads - 1) / threads);
        seg_base_kernel<<<blocks, threads, 0, stream>>>(idx_i, (int)n_total, base);
    }

    if (P <= 1024 && (P % 4) == 0) {
        long long QP    = P / 4;
        long long nquad = N * QP;
        int threads = 256;                       // 8 wave32 waves per workgroup
        int blocks  = (int)((nquad + threads - 1) / threads);
        if (QP == 124) {                         // K=32 fast path: const divisor
            triples_quad_kernel<124><<<blocks, threads, 0, stream>>>(
                base, out_i, out_j, out_k, (int)K, (int)P, (int)QP, (int)nquad);
        } else {
            triples_quad_kernel<0><<<blocks, threads, 0, stream>>>(
                base, out_i, out_j, out_k, (int)K, (int)P, (int)QP, (int)nquad);
        }
    } else {
        // Fully scalar fallback (P not quad-divisible or too large for LDS).
        long long total = N * P;
        int threads = 256;
        int blocks  = (int)((total + threads - 1) / threads);
        triples_rem_kernel<<<blocks, threads, 0, stream>>>(
            base, out_i, out_j, out_k, (int)K, (int)P, /*p_start=*/0, (int)N);
        return;
    }

    if (P % 4) {  // remainder triples (not hit for K=32)
        long long rem   = P % 4;
        long long start = P - rem;
        long long total = N * rem;
        int threads = 256;
        int blocks  = (int)((total + threads - 1) / threads);
        triples_rem_kernel<<<blocks, threads, 0, stream>>>(
            base, out_i, out_j, out_k, (int)K, (int)P, (int)start, (int)N);
    }
}